// RoPE_full_matrix_19086834663469
// MI455X (gfx1250) — compile-verified
//
#include <hip/hip_runtime.h>
#include <stdint.h>

// RoPE (full-matrix reference collapses to elementwise pair rotation).
// Purely HBM-bound: 16 MB traffic -> ~0.7 us at 23.3 TB/s on MI455X.
//
// CDNA5 path: stage the x stream through LDS with the async copy engine
// (global_load_async_to_lds_b128, ASYNCcnt) in a per-wave double buffer,
// software-pipelining DMA of token i+1 against compute of token i.
// One wave == one token: 32 lanes x float4 == 128 floats == 512 B.

typedef float v4f __attribute__((ext_vector_type(4)));

#define INV_2PI 0.15915494309189535f
// -log2(10000)/64  (freq_k = exp2(C_FREQ * k) = 10000^(-2k/128))
#define C_FREQ  -0.20762050593046014f

#define WAVES_PER_BLOCK 8
#define TOKENS_PER_WAVE 4

__global__ __launch_bounds__(256) void rope_full_matrix_async_kernel(
    const float* __restrict__ x,     // [B*S, 128]
    const int*   __restrict__ pos,   // [B*S]
    float*       __restrict__ y,     // [B*S, 128]
    int n_tokens)                    // B*S
{
    // Per-wave private double buffer: no cross-wave sharing -> no barriers;
    // ASYNCcnt is per-wave, in-order completion gives us the sync we need.
    __shared__ v4f stage[WAVES_PER_BLOCK][2][32];

    const int wave_in_blk = (int)(threadIdx.x >> 5);
    const int lane        = (int)(threadIdx.x & 31);
    const int wave_id     = (int)blockIdx.x * WAVES_PER_BLOCK + wave_in_blk;
    const int n_waves     = (int)gridDim.x * WAVES_PER_BLOCK;

    // Pair frequencies are per-lane constants (lane == quad index in token).
    const float k0 = (float)(2 * lane);
    const float w0 = __builtin_amdgcn_exp2f(C_FREQ * k0) * INV_2PI;
    const float w1 = __builtin_amdgcn_exp2f(C_FREQ * (k0 + 1.0f)) * INV_2PI;

    // LDS byte offsets of this lane's 16B slots (generic LDS ptr keeps the
    // byte offset in addr[31:0]).
    const uint32_t lds0 = (uint32_t)(uintptr_t)&stage[wave_in_blk][0][lane];
    const uint32_t lds1 = (uint32_t)(uintptr_t)&stage[wave_in_blk][1][lane];

    int token = wave_id;
    if (token >= n_tokens) return;   // wave-uniform

    // Byte offset of this lane's 16B chunk of token t: (t*128 + lane*4)*4.
    uint32_t byte_off = ((uint32_t)token * 128u + (uint32_t)lane * 4u) * 4u;
    const uint32_t stride_bytes = (uint32_t)n_waves * 512u;

    // Prologue: async-stage token0 into buffer 0 (ASYNCcnt -> 1).
    asm volatile("global_load_async_to_lds_b128 %0, %1, %2"
                 :: "v"(lds0), "v"(byte_off), "s"(x) : "memory");

    int cb = 0;                      // buffer holding the current token
    for (;;) {
        const int  next     = token + n_waves;          // wave-uniform
        const bool has_next = next < n_tokens;

        if (has_next) {
            // Kick the DMA for the next token into the other buffer while
            // we compute the current one.
            asm volatile("global_load_async_to_lds_b128 %0, %1, %2"
                         :: "v"(cb ? lds0 : lds1),
                            "v"(byte_off + stride_bytes), "s"(x) : "memory");
            // And warm L2 two tokens ahead (global_prefetch_b8).
            const int pf = token + 2 * n_waves;
            if (pf < n_tokens) {
                __builtin_prefetch(&x[(size_t)pf * 128 + (size_t)lane * 4], 0, 0);
            }
            // Current buffer resident once only the just-issued op remains.
            asm volatile("s_wait_asynccnt 0x1" ::: "memory");
        } else {
            asm volatile("s_wait_asynccnt 0x0" ::: "memory");
        }

        // ds_load_b128 from the staged buffer.
        const v4f   xv = stage[wave_in_blk][cb][lane];
        const float p  = (float)pos[token];

        // Angle in turns; reduce to [-0.5, 0.5] with v_rndne so hardware
        // v_sin/v_cos (which take turns) stay in their accurate domain.
        float t0 = p * w0;
        float t1 = p * w1;
        t0 -= __builtin_rintf(t0);
        t1 -= __builtin_rintf(t1);
        const float s0 = __builtin_amdgcn_sinf(t0);
        const float c0 = __builtin_amdgcn_cosf(t0);
        const float s1 = __builtin_amdgcn_sinf(t1);
        const float c1 = __builtin_amdgcn_cosf(t1);

        v4f yv;
        yv.x = c0 * xv.x - s0 * xv.y;   // even = cos*xe - sin*xo
        yv.y = s0 * xv.x + c0 * xv.y;   // odd  = sin*xe + cos*xo
        yv.z = c1 * xv.z - s1 * xv.w;
        yv.w = s1 * xv.z + c1 * xv.w;

        // Streaming single-use output: non-temporal B128 store.
        __builtin_nontemporal_store(
            yv, (v4f*)(y + (size_t)token * 128 + (size_t)lane * 4));

        if (!has_next) break;
        token     = next;
        byte_off += stride_bytes;
        cb       ^= 1;
    }
}

extern "C" void kernel_launch(void* const* d_in, const int* in_sizes, int n_in,
                              void* d_out, int out_size, void* d_ws, size_t ws_size,
                              hipStream_t stream) {
    (void)n_in; (void)out_size; (void)d_ws; (void)ws_size;

    const float* x   = (const float*)d_in[0];   // float32 [B,S,128]
    const int*   pos = (const int*)d_in[1];     // int32   [B,S]
    float*       y   = (float*)d_out;           // float32 [B,S,128]

    const int n_tokens = in_sizes[1];           // B*S

    // ~TOKENS_PER_WAVE tokens per wave so the async double buffer pipelines.
    int blocks = (n_tokens + WAVES_PER_BLOCK * TOKENS_PER_WAVE - 1) /
                 (WAVES_PER_BLOCK * TOKENS_PER_WAVE);
    if (blocks < 1) blocks = 1;

    rope_full_matrix_async_kernel<<<blocks, WAVES_PER_BLOCK * 32, 0, stream>>>(
        x, pos, y, n_tokens);
}